// MLA_18090402251202
// MI455X (gfx1250) — compile-verified
//
#include <hip/hip_runtime.h>
#include <hip/hip_bf16.h>

typedef __attribute__((ext_vector_type(2))) float v2f;
typedef __attribute__((ext_vector_type(8))) float v8f;

#define BM 64
#define BN 32
#define BK 16
#define LDSTR 18   // padded LDS stride: conflict-free 16-row bursts, pairs stay 8B-aligned

// CDNA5 async global->LDS copy (ASYNCcnt-tracked), per CDNA5 ISA.
// VDST operand = per-lane LDS byte offset (low 32 bits of generic pointer to __shared__).
#define ASYNC_LDS_B128(ldsoff, gptr)                                        \
    asm volatile("global_load_async_to_lds_b128 %0, %1, off"                \
                 :: "v"(ldsoff), "v"(gptr) : "memory")
#define ASYNC_LDS_B64(ldsoff, gptr)                                         \
    asm volatile("global_load_async_to_lds_b64 %0, %1, off"                 \
                 :: "v"(ldsoff), "v"(gptr) : "memory")
#define WAIT_ASYNC_2() asm volatile("s_wait_asynccnt 0x2" ::: "memory")
#define WAIT_ASYNC_0() asm volatile("s_wait_asynccnt 0x0" ::: "memory")

// C[m,n] = sum_k A[m,k] * W[n,k] + bias[n]
// SCATTER=false: dst[m*N + n]
// SCATTER=true : dst[((b*16+h)*4096 + s)*rowlen + colbase + d], b=m>>12, s=m&4095, h=n>>7, d=n&127
template <bool SCATTER>
__global__ __launch_bounds__(256) void gemm_f32_wmma(
    const float* __restrict__ A, const float* __restrict__ W,
    const float* __restrict__ bias, float* __restrict__ dst,
    int M, int N, int K, int rowlen, int colbase)
{
    __shared__ float Alds[2][BM * LDSTR];
    __shared__ float Wlds[2][BN * LDSTR];

    const int tid  = threadIdx.x;
    const int lane = tid & 31;
    const int wave = tid >> 5;     // 8 waves: 4 (M) x 2 (N)
    const int wm   = wave >> 1;
    const int wn   = wave & 1;

    const int m0 = blockIdx.y * BM;
    const int n0 = blockIdx.x * BN;

    // per-lane fragment addressing (V_WMMA_F32_16X16X4_F32 layout):
    // lanes 0-15 hold K=0,1 ; lanes 16-31 hold K=2,3
    const int halfsel = (lane >> 4) << 1;
    const int fa = (wm * 16 + (lane & 15)) * LDSTR;  // A lds row (M = lane&15)
    const int fb = (wn * 16 + (lane & 15)) * LDSTR;  // W lds row (N = lane&15)

    // cooperative async tile loads: every lane issues exactly one b128 (A) and
    // one b64 (W) per K-chunk so per-wave ASYNCcnt bookkeeping is uniform.
    const int ar  = tid >> 2;            // 0..63  A row
    const int ac  = (tid & 3) << 2;      // 0,4,8,12
    const int wr  = tid >> 3;            // 0..31  W row
    const int wc  = (tid & 7) << 1;      // 0,2,..,14

    const float* Ag = A + (size_t)(m0 + ar) * K + ac;
    const float* Wg = W + (size_t)(n0 + wr) * K + wc;

    unsigned ldsA[2], ldsW[2];
    ldsA[0] = (unsigned)(uintptr_t)&Alds[0][ar * LDSTR + ac];
    ldsA[1] = (unsigned)(uintptr_t)&Alds[1][ar * LDSTR + ac];
    ldsW[0] = (unsigned)(uintptr_t)&Wlds[0][wr * LDSTR + wc];
    ldsW[1] = (unsigned)(uintptr_t)&Wlds[1][wr * LDSTR + wc];

    v8f acc = {0.f, 0.f, 0.f, 0.f, 0.f, 0.f, 0.f, 0.f};

    auto compute = [&](const float* As, const float* Ws) {
#pragma unroll
        for (int kk = 0; kk < BK; kk += 4) {
            v2f af, bf;
            af.x = As[fa + kk + halfsel];
            af.y = As[fa + kk + halfsel + 1];
            bf.x = Ws[fb + kk + halfsel];
            bf.y = Ws[fb + kk + halfsel + 1];
            acc = __builtin_amdgcn_wmma_f32_16x16x4_f32(
                false, af, false, bf, (short)0, acc, false, false);
        }
    };

    const int T = K / BK;
    int cur = 0;

    // prologue: prefetch tile 0
    ASYNC_LDS_B128(ldsA[0], Ag);
    ASYNC_LDS_B64(ldsW[0], Wg);

    for (int t = 0; t < T - 1; ++t) {
        // prefetch tile t+1 into the other buffer while tile t is consumed
        const float* Agn = Ag + (size_t)(t + 1) * BK;
        const float* Wgn = Wg + (size_t)(t + 1) * BK;
        ASYNC_LDS_B128(ldsA[cur ^ 1], Agn);
        ASYNC_LDS_B64(ldsW[cur ^ 1], Wgn);
        WAIT_ASYNC_2();       // tile t's 2 async ops done (loads complete in order)
        __syncthreads();      // all waves' tile t data resident in LDS
        compute(Alds[cur], Wlds[cur]);
        __syncthreads();      // all waves done reading buf before it is re-filled
        cur ^= 1;
    }
    WAIT_ASYNC_0();
    __syncthreads();
    compute(Alds[cur], Wlds[cur]);

    // epilogue: bias + (optional) scatter into (B,H,S,*) concat layout
    const int gn = n0 + wn * 16 + (lane & 15);
    const float bv = bias[gn];
    const int rbase = m0 + wm * 16 + ((lane >> 4) << 3);  // M = g + 8*(lane>=16)
#pragma unroll
    for (int g = 0; g < 8; ++g) {
        const int gm = rbase + g;
        const float val = acc[g] + bv;
        if (SCATTER) {
            const int bb = gm >> 12;
            const int ss = gm & 4095;
            const int hh = gn >> 7;
            const int dd = gn & 127;
            const size_t idx =
                (size_t)((((bb << 4) + hh) << 12) + ss) * (size_t)rowlen + colbase + dd;
            dst[idx] = val;
        } else {
            dst[(size_t)gm * N + gn] = val;
        }
    }
}

// RoPE: Q[...,0:128] in place (pairs j, j+64); K[...,128:256] from stage-1 k_r workspace.
// angle = s * 10000^(-j/1024), exactly the reference tables.
__global__ __launch_bounds__(256) void rope_kernel(
    float* __restrict__ Q, float* __restrict__ Kout, const float* __restrict__ kr_ws)
{
    const long t = (long)blockIdx.x * blockDim.x + threadIdx.x;  // B*H*S*64 threads
    const int j = (int)(t & 63);
    const int s = (int)((t >> 6) & 4095);
    const int h = (int)((t >> 18) & 15);
    const int b = (int)(t >> 22);

    const float invf = expf((float)j * -0.0089944730195079925f);  // -ln(10000)/1024
    const float ang  = (float)s * invf;
    float sn, cs;
    sincosf(ang, &sn, &cs);

    const size_t base = (size_t)((((b << 4) + h) << 12) + s) * 256;

    // Q rope (in place on first 128 features)
    const float q1 = Q[base + j];
    const float q2 = Q[base + 64 + j];
    Q[base + j]      = q1 * cs - q2 * sn;
    Q[base + 64 + j] = q1 * sn + q2 * cs;

    // K rope tail, broadcast over heads from (m = b*4096+s, 128) workspace
    const size_t m = ((size_t)b << 12) + s;
    const float k1 = kr_ws[m * 128 + j];
    const float k2 = kr_ws[m * 128 + 64 + j];
    Kout[base + 128 + j] = k1 * cs - k2 * sn;
    Kout[base + 192 + j] = k1 * sn + k2 * cs;
}

extern "C" void kernel_launch(void* const* d_in, const int* in_sizes, int n_in,
                              void* d_out, int out_size, void* d_ws, size_t ws_size,
                              hipStream_t stream) {
    (void)in_sizes; (void)n_in; (void)out_size; (void)ws_size;

    const float* x       = (const float*)d_in[0];
    const float* W_dkv_w = (const float*)d_in[1];
    const float* W_dkv_b = (const float*)d_in[2];
    const float* W_uk_w  = (const float*)d_in[3];
    const float* W_uk_b  = (const float*)d_in[4];
    const float* W_kr_w  = (const float*)d_in[5];
    const float* W_kr_b  = (const float*)d_in[6];
    const float* W_uv_w  = (const float*)d_in[7];
    const float* W_uv_b  = (const float*)d_in[8];
    const float* W_dq_w  = (const float*)d_in[9];
    const float* W_dq_b  = (const float*)d_in[10];
    const float* W_uq_w  = (const float*)d_in[11];
    const float* W_uq_b  = (const float*)d_in[12];
    const float* W_qr_w  = (const float*)d_in[13];
    const float* W_qr_b  = (const float*)d_in[14];

    float* out = (float*)d_out;
    float* ws  = (float*)d_ws;

    const int M = 8192;                 // B*S
    const size_t QSZ = 33554432;        // 2*16*4096*256
    float* ws_cdkv = ws;                // 8192*128
    float* ws_kr   = ws + 1048576;      // 8192*128
    float* ws_cq   = ws + 2097152;      // 8192*2048

    const dim3 blk(256);
    const dim3 gN128(128 / BN, M / BM);
    const dim3 gN2048(2048 / BN, M / BM);

    // Stage 1: projections of x -> workspace
    gemm_f32_wmma<false><<<gN128, blk, 0, stream>>>(x, W_dkv_w, W_dkv_b, ws_cdkv, M, 128, 2048, 0, 0);
    gemm_f32_wmma<false><<<gN128, blk, 0, stream>>>(x, W_kr_w,  W_kr_b,  ws_kr,   M, 128, 2048, 0, 0);
    gemm_f32_wmma<false><<<gN2048, blk, 0, stream>>>(x, W_dq_w, W_dq_b,  ws_cq,   M, 2048, 2048, 0, 0);

    // Stage 2: up-projections scattered straight into (B,H,S,*) output slots
    gemm_f32_wmma<true><<<gN2048, blk, 0, stream>>>(ws_cdkv, W_uk_w, W_uk_b, out + QSZ,     M, 2048, 128,  256, 0);   // K_C
    gemm_f32_wmma<true><<<gN2048, blk, 0, stream>>>(ws_cdkv, W_uv_w, W_uv_b, out + 2 * QSZ, M, 2048, 128,  128, 0);   // V
    gemm_f32_wmma<true><<<gN2048, blk, 0, stream>>>(ws_cq,   W_uq_w, W_uq_b, out,           M, 2048, 2048, 256, 128); // q_c
    gemm_f32_wmma<true><<<gN2048, blk, 0, stream>>>(ws_cq,   W_qr_w, W_qr_b, out,           M, 2048, 2048, 256, 0);   // c_Qr (pre-rope)

    // Stage 3: RoPE (Q in place; K tail broadcast from k_r workspace)
    rope_kernel<<<(2 * 16 * 4096 * 64) / 256, blk, 0, stream>>>(out, out + QSZ, ws_kr);
}